// GraphSAGE_56255481643658
// MI455X (gfx1250) — compile-verified
//
#include <hip/hip_runtime.h>

typedef __attribute__((ext_vector_type(2))) float v2f;
typedef __attribute__((ext_vector_type(8))) float v8f;

__device__ __forceinline__ v8f wmma_f32_16x16x4(v2f a, v2f b, v8f c) {
  // 8-arg form: (neg_a, A, neg_b, B, c_mod, C, reuse_a, reuse_b)
  return __builtin_amdgcn_wmma_f32_16x16x4_f32(false, a, false, b, (short)0, c,
                                               false, false);
}

constexpr int EPI_NONE = 0;  // Y = X @ W
constexpr int EPI_SAGE = 1;  // Y = relu(X@W + agg/max(deg,1) + bias)
constexpr int EPI_BIAS = 2;  // Y = X @ W + bias

// Pack row-major W[K][NCOL] into pair-interleaved Wp[K/2][NCOL][2] so the WMMA
// B operand (W[k][n], W[k+1][n]) is one contiguous b64 load.
__global__ __launch_bounds__(256) void pack_w_kernel(const float* __restrict__ W,
                                                     float* __restrict__ Wp,
                                                     int K, int NCOL) {
  int i = blockIdx.x * 256 + threadIdx.x;
  if (i >= K * NCOL) return;
  int k = i / NCOL, n = i - k * NCOL;
  Wp[(size_t)(k >> 1) * NCOL * 2 + n * 2 + (k & 1)] = W[i];
}

// One wave computes MT 16-row tiles x (NT*16) cols of Y = X(N x KDIM) @ W.
// B loads (packed) are shared across the MT row tiles.
// 256 threads = 8 waves = 8*16*MT rows per block. Requires N % (16*MT) == 0
// (100000 = 3125 * 32).
template <int KDIM, int NT, int MT, int EPI>
__global__ __launch_bounds__(256) void gemm_wmma_kernel(
    const float* __restrict__ X, const float* __restrict__ Wp,
    const float* __restrict__ bias, const float* __restrict__ agg,
    const float* __restrict__ deg, float* __restrict__ Y, int N) {
  constexpr int NCOL = NT * 16;
  const int wave = blockIdx.x * 8 + (threadIdx.x >> 5);
  const int lane = threadIdx.x & 31;
  const int row0 = wave * (16 * MT);
  if (row0 >= N) return;  // wave-uniform exit: EXEC stays all-ones for WMMA
  const int half = lane >> 4;  // 0: K pair {k,k+1}, 1: K pair {k+2,k+3}
  const int m = lane & 15;     // A row / B,C column within 16-wide tile

  const float* xbase = X + (size_t)(row0 + m) * KDIM + 2 * half;
  v8f zero = {};
  v8f acc[MT][NT];
#pragma unroll
  for (int i = 0; i < MT; ++i)
#pragma unroll
    for (int j = 0; j < NT; ++j) acc[i][j] = zero;

#pragma unroll 2
  for (int k = 0; k < KDIM; k += 4) {
    v2f a[MT];
#pragma unroll
    for (int i = 0; i < MT; ++i) {
      a[i].x = xbase[(size_t)i * 16 * KDIM + k];
      a[i].y = xbase[(size_t)i * 16 * KDIM + k + 1];
    }
    // pair row (k + 2*half)/2 = k/2 + half
    const float* wp = Wp + ((size_t)(k / 2 + half) * NCOL + m) * 2;
#pragma unroll
    for (int j = 0; j < NT; ++j) {
      v2f b = *(const v2f*)(wp + j * 32);  // (W[k+2h][col], W[k+2h+1][col])
#pragma unroll
      for (int i = 0; i < MT; ++i) acc[i][j] = wmma_f32_16x16x4(a[i], b, acc[i][j]);
    }
  }

  // C/D layout: VGPR v -> row (v + 8*half), col = m (+ tile offset)
#pragma unroll
  for (int i = 0; i < MT; ++i) {
#pragma unroll
    for (int v = 0; v < 8; ++v) {
      const int r = row0 + i * 16 + v + 8 * half;
      float scale = 0.0f;
      if constexpr (EPI == EPI_SAGE) {
        scale = 1.0f / fmaxf(deg[r], 1.0f);
      }
#pragma unroll
      for (int j = 0; j < NT; ++j) {
        const int col = j * 16 + m;
        float val = acc[i][j][v];
        if constexpr (EPI == EPI_SAGE) {
          val = val + agg[(size_t)r * NCOL + col] * scale + bias[col];
          val = fmaxf(val, 0.0f);
        } else if constexpr (EPI == EPI_BIAS) {
          val = val + bias[col];
        }
        Y[(size_t)r * NCOL + col] = val;
      }
    }
  }
}

// deg[dst[e]] += 1
__global__ __launch_bounds__(256) void degree_kernel(const int* __restrict__ dst,
                                                     float* __restrict__ deg,
                                                     int E) {
  int e = blockIdx.x * 256 + threadIdx.x;
  if (e < E) unsafeAtomicAdd(&deg[dst[e]], 1.0f);
}

// One wave per edge; lane handles a float2 slice of the 64-float row.
// agg[dst[e]][:] += T[src[e]][:]   (reads/atomics stay L2-resident: hot set
// t+agg+deg ~ 52MB << 192MB L2)
__global__ __launch_bounds__(256) void scatter_kernel(const float* __restrict__ T,
                                                      const int* __restrict__ src,
                                                      const int* __restrict__ dst,
                                                      float* __restrict__ agg,
                                                      int E) {
  const unsigned tid = blockIdx.x * 256u + threadIdx.x;
  const int e = (int)(tid >> 5);
  if (e >= E) return;
  const int c = (int)(tid & 31u) * 2;
  const int s = src[e];
  const int d = dst[e];
  const float2 v = *(const float2*)(T + (size_t)s * 64 + c);
  float* ap = agg + (size_t)d * 64 + c;
  unsafeAtomicAdd(ap + 0, v.x);
  unsafeAtomicAdd(ap + 1, v.y);
}

extern "C" void kernel_launch(void* const* d_in, const int* in_sizes, int n_in,
                              void* d_out, int out_size, void* d_ws,
                              size_t ws_size, hipStream_t stream) {
  const float* x   = (const float*)d_in[0];
  const int*   ei  = (const int*)d_in[1];
  const float* W1l = (const float*)d_in[2];
  const float* b1  = (const float*)d_in[3];
  const float* W1r = (const float*)d_in[4];
  const float* W2l = (const float*)d_in[5];
  const float* b2  = (const float*)d_in[6];
  const float* W2r = (const float*)d_in[7];
  const float* Wo  = (const float*)d_in[8];
  const float* bo  = (const float*)d_in[9];
  float* out = (float*)d_out;

  const int N = in_sizes[0] / 128;  // 100000
  const int E = in_sizes[1] / 2;    // 1600000
  const int* src = ei;
  const int* dst = ei + E;

  // Workspace: t (N*64) | agg (N*64) | h1 (N*64) | deg (N) | packed weights
  float* t   = (float*)d_ws;
  float* agg = t + (size_t)N * 64;
  float* h1  = agg + (size_t)N * 64;
  float* deg = h1 + (size_t)N * 64;
  float* W1l_p = deg + N;            // 128*64
  float* W1r_p = W1l_p + 128 * 64;   // 128*64
  float* W2l_p = W1r_p + 128 * 64;   // 64*64
  float* W2r_p = W2l_p + 64 * 64;    // 64*64
  float* Wo_p  = W2r_p + 64 * 64;    // 64*112

  const int gemm_blocks = (N + 255) / 256;  // 8 waves * 32 rows per block
  const int sc_blocks   = (int)(((long long)E * 32 + 255) / 256);
  const int dg_blocks   = (E + 255) / 256;

  // Pack weights into pair-interleaved layout (trivial cost, <128KB total)
  pack_w_kernel<<<(128 * 64 + 255) / 256, 256, 0, stream>>>(W1l, W1l_p, 128, 64);
  pack_w_kernel<<<(128 * 64 + 255) / 256, 256, 0, stream>>>(W1r, W1r_p, 128, 64);
  pack_w_kernel<<<(64 * 64 + 255) / 256, 256, 0, stream>>>(W2l, W2l_p, 64, 64);
  pack_w_kernel<<<(64 * 64 + 255) / 256, 256, 0, stream>>>(W2r, W2r_p, 64, 64);
  pack_w_kernel<<<(64 * 112 + 255) / 256, 256, 0, stream>>>(Wo, Wo_p, 64, 112);

  // Degree (shared by both layers)
  hipMemsetAsync(deg, 0, (size_t)N * sizeof(float), stream);
  degree_kernel<<<dg_blocks, 256, 0, stream>>>(dst, deg, E);

  // ---- Layer 1: h1 = relu(mean(x)@W1l + b1 + x@W1r)
  // linearity: mean(x)@W1l == segsum((x@W1l)[src]) / deg  -> aggregate 64 dims
  gemm_wmma_kernel<128, 4, 2, EPI_NONE><<<gemm_blocks, 256, 0, stream>>>(
      x, W1l_p, nullptr, nullptr, nullptr, t, N);
  hipMemsetAsync(agg, 0, (size_t)N * 64 * sizeof(float), stream);
  scatter_kernel<<<sc_blocks, 256, 0, stream>>>(t, src, dst, agg, E);
  gemm_wmma_kernel<128, 4, 2, EPI_SAGE><<<gemm_blocks, 256, 0, stream>>>(
      x, W1r_p, b1, agg, deg, h1, N);

  // ---- Layer 2: h2 = relu(mean(h1)@W2l + b2 + h1@W2r)   (h2 reuses t)
  gemm_wmma_kernel<64, 4, 2, EPI_NONE><<<gemm_blocks, 256, 0, stream>>>(
      h1, W2l_p, nullptr, nullptr, nullptr, t, N);
  hipMemsetAsync(agg, 0, (size_t)N * 64 * sizeof(float), stream);
  scatter_kernel<<<sc_blocks, 256, 0, stream>>>(t, src, dst, agg, E);
  gemm_wmma_kernel<64, 4, 2, EPI_SAGE><<<gemm_blocks, 256, 0, stream>>>(
      h1, W2r_p, b2, agg, deg, t, N);  // t now holds h2

  // ---- Output: out = h2 @ Wo + bo   (112 = 7 x 16 column tiles)
  gemm_wmma_kernel<64, 7, 2, EPI_BIAS><<<gemm_blocks, 256, 0, stream>>>(
      t, Wo_p, bo, nullptr, nullptr, out, N);
}